// admm_video_22213570855381
// MI455X (gfx1250) — compile-verified
//
#include <hip/hip_runtime.h>

// ADMM video closed-form kernel for MI455X (gfx1250), v2: float4 / b128 path.
//
// Per-pixel math (reference collapses to this exactly; CG just converges to it):
//   m_f = (ce_f > 0), n_f = 1/8 - m_f
//   b0 = m.frames, b1 = n.frames              (noise ~5e-10 is sub-fp32-eps)
//   A  = rho*I + m m^T + n n^T  (8x8, rank-2 + identity)
//   two Woodbury 2x2 solves ->  out_f = x0_f + (m_f*dy0 + n_f*dy1),
//   x0_f = (f<4) ? b0 : b1.
//
// Memory-bound (~5.3 MB total). Each thread owns 4 consecutive pixels:
// async global->LDS staging in 16-byte (b128) transactions on the gfx1250
// ASYNCcnt path, b128 LDS reads, b128 stores. 225 blocks x 64 threads x 4 px.

#define FRAME_N 8
#define PH 180
#define PW 320
#define NP (PH * PW)   // 57600
#define BLK 64         // 2 wave32 per block
#define QPT 4          // pixels (floats) per thread
#define NQ (NP / QPT)  // 14400 quads = 225 * 64 exactly

#if __has_builtin(__builtin_amdgcn_global_load_async_to_lds_b128) && \
    __has_builtin(__builtin_amdgcn_s_wait_asynccnt)
#define USE_ASYNC_LDS 1
#else
#define USE_ASYNC_LDS 0
#endif

typedef int v4i __attribute__((vector_size(4 * sizeof(int))));
typedef __attribute__((address_space(1))) v4i gv4i_t;  // global int4
typedef __attribute__((address_space(3))) v4i lv4i_t;  // LDS int4

union F4 { float4 v; float a[QPT]; };

__global__ __launch_bounds__(BLK) void admm_video_closed_form(
    const float* __restrict__ frames,   // (8,180,320)
    const float* __restrict__ ce,       // (8,180,320)
    const float* __restrict__ rho_ptr,  // (1,)
    float* __restrict__ out)            // (8,180,320)
{
    __shared__ float4 s_fr[FRAME_N][BLK];
    __shared__ float4 s_ce[FRAME_N][BLK];

    const int tid = threadIdx.x;
    const int q   = blockIdx.x * BLK + tid;   // quad index; grid covers NQ exactly
    const size_t e0 = (size_t)q * QPT;        // first pixel of this thread's quad

#if USE_ASYNC_LDS
    // 16-byte async global->LDS copies; each lane's loads land in that lane's
    // own LDS slots, so a per-wave s_wait_asynccnt(0) is the only sync needed.
    #pragma unroll
    for (int f = 0; f < FRAME_N; ++f) {
        __builtin_amdgcn_global_load_async_to_lds_b128(
            (gv4i_t*)(frames + (size_t)f * NP + e0),
            (lv4i_t*)&s_fr[f][tid], 0, 0);
        __builtin_amdgcn_global_load_async_to_lds_b128(
            (gv4i_t*)(ce + (size_t)f * NP + e0),
            (lv4i_t*)&s_ce[f][tid], 0, 0);
    }
    const float rho = rho_ptr[0];        // overlaps with the async copies
    __builtin_amdgcn_s_wait_asynccnt(0);
#else
    #pragma unroll
    for (int f = 0; f < FRAME_N; ++f) {
        s_fr[f][tid] = *(const float4*)(frames + (size_t)f * NP + e0);
        s_ce[f][tid] = *(const float4*)(ce + (size_t)f * NP + e0);
    }
    const float rho = rho_ptr[0];
#endif

    // Mask bits and RHS reductions per pixel: b0 = m.frames, b1 = (1/8-m).frames
    unsigned mbits[QPT] = {0u, 0u, 0u, 0u};
    float b0[QPT] = {0.f, 0.f, 0.f, 0.f};
    float b1[QPT] = {0.f, 0.f, 0.f, 0.f};
    #pragma unroll
    for (int f = 0; f < FRAME_N; ++f) {
        F4 fv, cv;
        fv.v = s_fr[f][tid];
        cv.v = s_ce[f][tid];
        #pragma unroll
        for (int c = 0; c < QPT; ++c) {
            const bool  on = (cv.a[c] > 0.0f);
            const float mf = on ? 1.0f : 0.0f;
            mbits[c] |= (on ? 1u : 0u) << f;
            b0[c] = fmaf(mf, fv.a[c], b0[c]);
            b1[c] = fmaf(0.125f - mf, fv.a[c], b1[c]);
        }
    }

    const float inv_rho = 1.0f / rho;
    float dy0[QPT], dy1[QPT];
    #pragma unroll
    for (int c = 0; c < QPT; ++c) {
        const float kf   = (float)__builtin_popcount(mbits[c]);
        const float k4lo = (float)__builtin_popcount(mbits[c] & 0xFu); // frames 0..3

        // Gram matrix of U = [m n]
        const float mm = kf;
        const float mn = -0.875f * kf;
        const float nn = 0.125f * fmaf(6.0f, kf, 1.0f);

        const float M00 = rho + mm;
        const float M11 = rho + nn;
        const float M01 = mn;
        const float inv_det = 1.0f / fmaf(M00, M11, -(M01 * M01));

        // Solve 1: y = (rho*I2 + G)^-1 U^T (m b0 + n b1)
        const float t0 = fmaf(b0[c], mm, b1[c] * mn);
        const float t1 = fmaf(b0[c], mn, b1[c] * nn);
        const float y0 = inv_det * fmaf(M11, t0, -(M01 * t1));
        const float y1 = inv_det * fmaf(M00, t1, -(M01 * t0));

        // Solve 2: c2 = m y0 + n y1 + rho*x0, x0_f = (f<4) ? b0 : b1
        const float mx0 = fmaf(b0[c], k4lo, b1[c] * (kf - k4lo));   // m.x0
        const float nx0 = 0.5f * (b0[c] + b1[c]) - mx0;             // n.x0
        const float t0p = fmaf(y0, mm, fmaf(y1, mn, rho * mx0));
        const float t1p = fmaf(y0, mn, fmaf(y1, nn, rho * nx0));
        const float y0p = inv_det * fmaf(M11, t0p, -(M01 * t1p));
        const float y1p = inv_det * fmaf(M00, t1p, -(M01 * t0p));

        dy0[c] = (y0 - y0p) * inv_rho;
        dy1[c] = (y1 - y1p) * inv_rho;
    }

    // out_f = x0_f + m_f*dy0 + n_f*dy1, written as b128 per frame
    #pragma unroll
    for (int f = 0; f < FRAME_N; ++f) {
        F4 o;
        #pragma unroll
        for (int c = 0; c < QPT; ++c) {
            const float mf  = ((mbits[c] >> f) & 1u) ? 1.0f : 0.0f;
            const float nf  = 0.125f - mf;
            const float x0f = (f < 4) ? b0[c] : b1[c];
            o.a[c] = x0f + fmaf(mf, dy0[c], nf * dy1[c]);
        }
        *(float4*)(out + (size_t)f * NP + e0) = o.v;
    }
}

extern "C" void kernel_launch(void* const* d_in, const int* in_sizes, int n_in,
                              void* d_out, int out_size, void* d_ws, size_t ws_size,
                              hipStream_t stream) {
    (void)in_sizes; (void)n_in; (void)d_ws; (void)ws_size; (void)out_size;
    const float* frames = (const float*)d_in[0];  // (8,180,320) f32
    const float* ce     = (const float*)d_in[1];  // (8,180,320) f32
    const float* rho    = (const float*)d_in[2];  // (1,) f32
    float* out = (float*)d_out;                   // (8,180,320) f32

    const int blocks = NQ / BLK;  // 225, exact
    admm_video_closed_form<<<blocks, BLK, 0, stream>>>(frames, ce, rho, out);
}